// DescMatchingModule_54614804136390
// MI455X (gfx1250) — compile-verified
//
#include <hip/hip_runtime.h>
#include <hip/hip_bf16.h>
#include <math.h>

// Problem constants (from reference: b=4, c=128, h=w=24)
#define BATCH   4
#define CHANS   128
#define NPIX    576          // 24*24
#define NTILE   36           // 576/16
#define TILES_PER_B (NTILE * NTILE)      // 1296
#define TOTAL_TILES (BATCH * TILES_PER_B) // 5184
#define OUT_NORM_OFF (BATCH * NPIX * NPIX * 2)  // 2654208 floats

typedef __attribute__((ext_vector_type(2))) float v2f;
typedef __attribute__((ext_vector_type(8))) float v8f;

#define EPSN 1e-6f

// ---------------------------------------------------------------------------
// Kernel 1: per-descriptor sum-of-squares + inverse (eps + norm) factors.
// ws layout (floats): [0,2304) ss1 | [2304,4608) inv1 | [4608,6912) ss2 | [6912,9216) inv2
// ---------------------------------------------------------------------------
__global__ __launch_bounds__(256)
void descmatch_norms_kernel(const float* __restrict__ o1,
                            const float* __restrict__ o2,
                            float* __restrict__ ws) {
    int t = blockIdx.x * blockDim.x + threadIdx.x;
    if (t >= BATCH * NPIX) return;
    int b = t / NPIX;
    int i = t - b * NPIX;
    const float* p1 = o1 + (size_t)b * CHANS * NPIX + i;
    const float* p2 = o2 + (size_t)b * CHANS * NPIX + i;
    float s1 = 0.f, s2 = 0.f;
#pragma unroll 8
    for (int c = 0; c < CHANS; ++c) {
        float x = p1[(size_t)c * NPIX];
        float y = p2[(size_t)c * NPIX];
        s1 = fmaf(x, x, s1);
        s2 = fmaf(y, y, s2);
    }
    ws[t]                    = s1;
    ws[BATCH * NPIX + t]     = 1.f / (EPSN + sqrtf(s1));
    ws[2 * BATCH * NPIX + t] = s2;
    ws[3 * BATCH * NPIX + t] = 1.f / (EPSN + sqrtf(s2));
}

// ---------------------------------------------------------------------------
// Kernel 2: fused triple-GEMM per 16x16 tile using V_WMMA_F32_16X16X4_F32.
//   accC = d1 * d2^T                (for the distance matrix)
//   acc0 = (d1 .* W[0]) * d2^T     (fc output channel 0)
//   acc1 = (d1 .* W[1]) * d2^T     (fc output channel 1)
// One wave (32 lanes) per 16x16 output tile. EXEC all-ones, no divergence.
// ---------------------------------------------------------------------------
__global__ __launch_bounds__(256)
void descmatch_wmma_kernel(const float* __restrict__ o1,
                           const float* __restrict__ o2,
                           const float* __restrict__ W,
                           const float* __restrict__ bias,
                           const float* __restrict__ ws,
                           float* __restrict__ dout) {
    const int lane = threadIdx.x & 31;
    const int wid  = threadIdx.x >> 5;
    const int gt   = blockIdx.x * 8 + wid;           // global tile id, exact cover

    const int b  = gt / TILES_PER_B;
    const int r  = gt - b * TILES_PER_B;
    const int ti = r / NTILE;
    const int tj = r - ti * NTILE;
    const int i0 = ti * 16;
    const int j0 = tj * 16;

    const int half = lane >> 4;        // 0: K pair {0,1}; 1: K pair {2,3}
    const int lm   = lane & 15;        // row (A) / col (B) within tile
    const int koff = half * 2;

    // Memory is [b][c][pixel]; pixel contiguous => coalesced across lanes.
    const float* Ap = o1 + (size_t)b * CHANS * NPIX + i0 + lm;
    const float* Bp = o2 + (size_t)b * CHANS * NPIX + j0 + lm;

    v8f accC = {};
    v8f acc0 = {};
    v8f acc1 = {};

#pragma unroll 4
    for (int c0 = 0; c0 < CHANS; c0 += 4) {
        const int c = c0 + koff;
        // A fragment: lane holds A[lm][k] for k = koff, koff+1
        float a0 = Ap[(size_t)c * NPIX];
        float a1 = Ap[(size_t)(c + 1) * NPIX];
        // B fragment: lane holds B[k][lm] = d2[j0+lm][c..c+1]
        float b0 = Bp[(size_t)c * NPIX];
        float b1 = Bp[(size_t)(c + 1) * NPIX];
        // W pairs (c is even): broadcast loads
        float2 w0 = *(const float2*)(W + c);
        float2 w1 = *(const float2*)(W + CHANS + c);

        v2f av  = {a0, a1};
        v2f bv  = {b0, b1};
        v2f aw0 = {a0 * w0.x, a1 * w0.y};
        v2f aw1 = {a0 * w1.x, a1 * w1.y};

        // D = A x B + C   (f32, 16x16x4)
        accC = __builtin_amdgcn_wmma_f32_16x16x4_f32(
            false, av,  false, bv, (short)0, accC, false, false);
        acc0 = __builtin_amdgcn_wmma_f32_16x16x4_f32(
            false, aw0, false, bv, (short)0, acc0, false, false);
        acc1 = __builtin_amdgcn_wmma_f32_16x16x4_f32(
            false, aw1, false, bv, (short)0, acc1, false, false);
    }

    // Epilogue. C/D layout: lane -> col j0+(lane&15); VGPR v -> row i0+v+8*half.
    const float* ssA  = ws;
    const float* invA = ws + BATCH * NPIX;
    const float* ssB  = ws + 2 * BATCH * NPIX;
    const float* invB = ws + 3 * BATCH * NPIX;

    const int j = j0 + lm;
    const float ss2v = ssB[b * NPIX + j];
    const float in2  = invB[b * NPIX + j];
    const float b2   = ss2v * in2 * in2;
    const float bi0  = bias[0];
    const float bi1  = bias[1];

    float* outp = dout;                 // [b, i, j, 2] flattened
    float* onp  = dout + OUT_NORM_OFF;  // [b, i, j]    flattened

#pragma unroll
    for (int v = 0; v < 8; ++v) {
        const int i = i0 + v + half * 8;
        const float ss1v = ssA[b * NPIX + i];
        const float in1  = invA[b * NPIX + i];
        const float a2   = ss1v * in1 * in1;

        const float crossn = accC[v] * in1 * in2;
        float sq = a2 + b2 - 2.0f * crossn;
        sq = sq < 0.f ? 0.f : sq;

        const size_t flat = ((size_t)b * NPIX + i) * NPIX + j;
        onp[flat] = sqrtf(sq);

        float2 o2v;
        o2v.x = acc0[v] + bi0;
        o2v.y = acc1[v] + bi1;
        *(float2*)(outp + flat * 2) = o2v;
    }
}

extern "C" void kernel_launch(void* const* d_in, const int* in_sizes, int n_in,
                              void* d_out, int out_size, void* d_ws, size_t ws_size,
                              hipStream_t stream) {
    (void)in_sizes; (void)n_in; (void)out_size; (void)ws_size;
    const float* o1   = (const float*)d_in[0];
    const float* o2   = (const float*)d_in[1];
    const float* W    = (const float*)d_in[2];
    const float* bias = (const float*)d_in[3];
    float* ws  = (float*)d_ws;
    float* out = (float*)d_out;

    // 1) per-row norms into workspace (36 KB)
    {
        const int nthreads = BATCH * NPIX;              // 2304
        dim3 grid((nthreads + 255) / 256), block(256);
        descmatch_norms_kernel<<<grid, block, 0, stream>>>(o1, o2, ws);
    }
    // 2) fused WMMA triple-GEMM + epilogue; 5184 tiles, 8 waves/block
    {
        dim3 grid(TOTAL_TILES / 8), block(256);         // 648 blocks
        descmatch_wmma_kernel<<<grid, block, 0, stream>>>(o1, o2, W, bias, ws, out);
    }
}